// ChebConv_38809324486706
// MI455X (gfx1250) — compile-verified
//
#include <hip/hip_runtime.h>

typedef __attribute__((ext_vector_type(2))) float v2f;
typedef __attribute__((ext_vector_type(8))) float v8f;

#define NNODES 50000
#define NEDGES 1600000
#define CH 256

// ---------------------------------------------------------------------------
// Zero a float buffer (workspace Z accumulators).
__global__ void zero_kernel(float* __restrict__ p, long n) {
    long i = (long)blockIdx.x * blockDim.x + threadIdx.x;
    long stride = (long)gridDim.x * blockDim.x;
    for (; i < n; i += stride) p[i] = 0.0f;
}

// ---------------------------------------------------------------------------
// out[real; imag] = broadcast bias (channel = i % 256).
__global__ void init_out_kernel(float* __restrict__ out, const float* __restrict__ bias, long n) {
    long i = (long)blockIdx.x * blockDim.x + threadIdx.x;
    long stride = (long)gridDim.x * blockDim.x;
    for (; i < n; i += stride) out[i] = bias[i & (CH - 1)];
}

// ---------------------------------------------------------------------------
// SpMM scatter phase for one hop k:
//   Zr[row,:] += vr*Xr[col,:] - vi*Xi[col,:]
//   Zi[row,:] += vi*Xr[col,:] + vr*Xi[col,:]
// One 256-thread block handles EPB edges; lane c owns channel c.
// Gathers hit L2 (X fits in 192MB L2); scatters use global_atomic_add_f32.
__global__ __launch_bounds__(256) void spmm_kernel(
    const float* __restrict__ Xr, const float* __restrict__ Xi,
    const float* __restrict__ Lr, const float* __restrict__ Li,
    const int* __restrict__ rows, const int* __restrict__ cols,
    float* __restrict__ Zr, float* __restrict__ Zi,
    int nEdges, int edgesPerBlock)
{
    const int c = threadIdx.x;                       // channel 0..255
    long e0 = (long)blockIdx.x * edgesPerBlock;
    for (int i = 0; i < edgesPerBlock; ++i) {
        long e = e0 + i;
        if (e >= nEdges) return;
        const int row = rows[e];                     // uniform -> s_load
        const int col = cols[e];
        const float vr = Lr[e];
        const float vi = Li[e];
        const float xr = Xr[(long)col * CH + c];
        const float xi = Xi[(long)col * CH + c];
        atomicAdd(&Zr[(long)row * CH + c], vr * xr - vi * xi);
        atomicAdd(&Zi[(long)row * CH + c], vi * xr + vr * xi);
    }
}

// ---------------------------------------------------------------------------
// Dense accumulate-GEMM: out += Z @ W   (Z: [N,256], W: [256,256], fp32)
// One wave (32 threads) computes a 16(M) x 64(N) tile: 4 accumulators share
// one A-fragment per K-step of 4, using V_WMMA_F32_16X16X4_F32.
//
// A-frag (16x4 f32, 2 VGPRs): lanes 0-15 -> M=lane, K={k0,k0+1};
//                             lanes 16-31 -> M=lane-16, K={k0+2,k0+3}.
// B-frag (4x16 f32, 2 VGPRs): lanes 0-15 -> N=lane, K={k0,k0+1};
//                             lanes 16-31 -> N=lane-16, K={k0+2,k0+3}.
// C/D (16x16 f32, 8 VGPRs):   VGPR r -> M=r (lanes 0-15), M=r+8 (lanes 16-31).
__global__ __launch_bounds__(32) void gemm_wmma_kernel(
    const float* __restrict__ Zr, const float* __restrict__ Zi,
    const float* __restrict__ W,
    float* __restrict__ outR, float* __restrict__ outI)
{
    const int lane = threadIdx.x & 31;
    const int half = lane >> 4;       // 0: lanes 0-15, 1: lanes 16-31
    const int l    = lane & 15;
    const long rowBase = (long)blockIdx.x * 16;
    const int  colBase = blockIdx.y * 64;

    const float* __restrict__ Z   = (blockIdx.z == 0) ? Zr : Zi;
    float* __restrict__       out = (blockIdx.z == 0) ? outR : outI;

    const float* zrow = Z + (rowBase + l) * CH;   // A: row per lane
    const float* wcol = W + colBase + l;          // B: col per lane

    v8f acc0 = {}; v8f acc1 = {}; v8f acc2 = {}; v8f acc3 = {};

#pragma unroll 4
    for (int k0 = 0; k0 < CH; k0 += 4) {
        const int ka = k0 + 2 * half;
        // A fragment: 8-byte aligned v2f load (global_load_b64)
        v2f a = *(const v2f*)(zrow + ka);

        v2f b0, b1, b2, b3;
        b0.x = wcol[(long)ka * CH +  0]; b0.y = wcol[(long)(ka + 1) * CH +  0];
        b1.x = wcol[(long)ka * CH + 16]; b1.y = wcol[(long)(ka + 1) * CH + 16];
        b2.x = wcol[(long)ka * CH + 32]; b2.y = wcol[(long)(ka + 1) * CH + 32];
        b3.x = wcol[(long)ka * CH + 48]; b3.y = wcol[(long)(ka + 1) * CH + 48];

        acc0 = __builtin_amdgcn_wmma_f32_16x16x4_f32(false, a, false, b0, (short)0, acc0, false, false);
        acc1 = __builtin_amdgcn_wmma_f32_16x16x4_f32(false, a, false, b1, (short)0, acc1, false, false);
        acc2 = __builtin_amdgcn_wmma_f32_16x16x4_f32(false, a, false, b2, (short)0, acc2, false, false);
        acc3 = __builtin_amdgcn_wmma_f32_16x16x4_f32(false, a, false, b3, (short)0, acc3, false, false);
    }

    // Epilogue: tiles are disjoint per wave within a launch; launches serialize
    // on the stream, so plain read-add-write is race-free.
    const int rbase = 8 * half;
#pragma unroll
    for (int r = 0; r < 8; ++r) {
        const long row = rowBase + rbase + r;
        float* o = out + row * CH + colBase + l;
        o[ 0] += acc0[r];
        o[16] += acc1[r];
        o[32] += acc2[r];
        o[48] += acc3[r];
    }
}

// ---------------------------------------------------------------------------
extern "C" void kernel_launch(void* const* d_in, const int* in_sizes, int n_in,
                              void* d_out, int out_size, void* d_ws, size_t ws_size,
                              hipStream_t stream) {
    const float* Xr   = (const float*)d_in[0];   // [N,256]
    const float* Xi   = (const float*)d_in[1];   // [N,256]
    const float* Lr   = (const float*)d_in[2];   // [3,E]
    const float* Li   = (const float*)d_in[3];   // [3,E]
    const float* W    = (const float*)d_in[4];   // [3,256,256]
    const float* bias = (const float*)d_in[5];   // [1,256]
    const int*   rows = (const int*)d_in[6];     // [E]
    const int*   cols = (const int*)d_in[7];     // [E]

    float* out  = (float*)d_out;
    float* outR = out;
    float* outI = out + (long)NNODES * CH;

    float* Zr = (float*)d_ws;                    // [N,256] fp32
    float* Zi = Zr + (long)NNODES * CH;          // [N,256] fp32 (102.4 MB total)

    const long outN = 2L * NNODES * CH;
    init_out_kernel<<<2048, 256, 0, stream>>>(out, bias, outN);

    const int EPB = 8;
    const int spmmBlocks = (NEDGES + EPB - 1) / EPB;
    dim3 gemmGrid(NNODES / 16, CH / 64, 2);      // 3125 row tiles x 4 col tiles x {re,im}

    for (int k = 0; k < 3; ++k) {
        zero_kernel<<<2048, 256, 0, stream>>>(Zr, 2L * NNODES * CH);
        spmm_kernel<<<spmmBlocks, 256, 0, stream>>>(
            Xr, Xi, Lr + (long)k * NEDGES, Li + (long)k * NEDGES,
            rows, cols, Zr, Zi, NEDGES, EPB);
        gemm_wmma_kernel<<<gemmGrid, 32, 0, stream>>>(
            Zr, Zi, W + (long)k * CH * CH, outR, outI);
    }
}